// my_grid_linear_79783312490826
// MI455X (gfx1250) — compile-verified
//
#include <hip/hip_runtime.h>
#include <stdint.h>

typedef unsigned int       u32;
typedef unsigned long long u64;
typedef unsigned int u32x4 __attribute__((ext_vector_type(4)));
typedef int          i32x8 __attribute__((ext_vector_type(8)));
typedef int          i32x4 __attribute__((ext_vector_type(4)));
typedef float        f32x2 __attribute__((ext_vector_type(2)));
typedef float        f32x4 __attribute__((ext_vector_type(4)));

#define PLANE   (512 * 512)   // elements per (level,feature) plane
#define NLEVELS 16
#define KLDS    6             // levels staged in LDS
#define SMEM_BYTES 57344      // 57288 rounded up

// N_l = int(16 * 1.26^l)  (float64 math then trunc, matches numpy)
#define N_LIST {16,20,25,32,40,50,64,80,101,128,161,203,256,322,406,512}

// ---------------------------------------------------------------------------
// Kernel 1: repack grid_table [l][f][y][x] -> ws [l][y][x][f]  (float2).
// gt is read once: non-temporal loads. tbl stores stay RT so the 32MB table
// rinses into the 192MB L2 for the sample kernel's gathers.
// ---------------------------------------------------------------------------
__global__ __launch_bounds__(256) void mrgrid_repack(const float* __restrict__ gt,
                                                     f32x2* __restrict__ tbl) {
  int idx = blockIdx.x * 256 + threadIdx.x;           // 16 * 262144 threads
  int l   = idx >> 18;
  int rem = idx & (PLANE - 1);
  float f0 = __builtin_nontemporal_load(gt + (size_t)(2 * l + 0) * PLANE + rem);
  float f1 = __builtin_nontemporal_load(gt + (size_t)(2 * l + 1) * PLANE + rem);
  f32x2 v = {f0, f1};
  tbl[idx] = v;
}

// ---------------------------------------------------------------------------
// Kernel 2: sample. Levels 0..5 from LDS (TDM-staged patches), 6..14 from the
// L2-resident repacked table (interior, no masks), 15 masked (zeros padding).
// Output is write-once: non-temporal stores keep L2 free for the table.
// ---------------------------------------------------------------------------
__global__ __launch_bounds__(256) void mrgrid_sample(const f32x2* __restrict__ x,
                                                     const f32x2* __restrict__ tbl,
                                                     float* __restrict__ out,
                                                     int nPoints) {
  extern __shared__ float smem[];
  constexpr int NL[NLEVELS] = N_LIST;
  constexpr int P0[KLDS]   = {247, 245, 243, 239, 235, 230};   // (511-N)>>1
  constexpr int PW[KLDS]   = {18, 22, 27, 34, 42, 52};         // N+2
  constexpr int LOFS[KLDS] = {0, 2592, 6464, 12296, 21544, 35656}; // bytes

#if defined(__gfx1250__) && __has_builtin(__builtin_amdgcn_tensor_load_to_lds)
  if (threadIdx.x < 32) {   // one wave issues all TDM loads
    u32 ldsBase = (u32)(uintptr_t)smem;   // low 32 bits of LDS aperture addr
#pragma unroll
    for (int l = 0; l < KLDS; ++l) {
      u64 gaddr = (u64)(uintptr_t)(tbl + (size_t)l * PLANE + P0[l] * 512 + P0[l]);
      u32x4 g0;
      g0[0] = 1u;                                  // count=1, user mode
      g0[1] = ldsBase + (u32)LOFS[l];              // lds_addr (bytes)
      g0[2] = (u32)gaddr;                          // global_addr[31:0]
      g0[3] = ((u32)(gaddr >> 32) & 0x01FFFFFFu) | 0x80000000u; // [56:32] | type=2
      i32x8 g1;
      g1[0] = 0x00020000;                          // data_size=2 (4B), no multicast
      g1[1] = 0;                                   // abar=0, tensor_dim0 lo16 = 0
      g1[2] = 0x10;                                // tensor_dim0 = 0x100000 (never OOB)
      g1[3] = 0x10 | ((PW[l] * 2) << 16);          // tensor_dim1 hi, tile_dim0 = W*2 floats
      g1[4] = PW[l];                               // tile_dim1 = W rows, tile_dim2 unused
      g1[5] = 1024;                                // tensor_dim0_stride = 1024 floats
      g1[6] = 0;
      g1[7] = 0;
      i32x4 z4 = {0, 0, 0, 0};                     // groups 2/3: dims unused
      i32x8 z8 = {0, 0, 0, 0, 0, 0, 0, 0};         // extra group (clang-23 6-arg form)
      __builtin_amdgcn_tensor_load_to_lds(g0, g1, z4, z4, z8, 0);
    }
    __builtin_amdgcn_s_wait_tensorcnt(0);
  }
  __syncthreads();
#else
  // cooperative fallback staging
  for (int l = 0; l < KLDS; ++l) {
    int w = PW[l];
    f32x2* dst = (f32x2*)((char*)smem + LOFS[l]);
    const f32x2* src = tbl + (size_t)l * PLANE + P0[l] * 512 + P0[l];
    for (int i = threadIdx.x; i < w * w; i += blockDim.x)
      dst[i] = src[(i / w) * 512 + (i % w)];
  }
  __syncthreads();
#endif

  int stride = gridDim.x * blockDim.x;
  for (int b = blockIdx.x * blockDim.x + threadIdx.x; b < nPoints; b += stride) {
    f32x2 c = __builtin_nontemporal_load(x + b);   // read-once input
    float acc0[NLEVELS], acc1[NLEVELS];
#pragma unroll
    for (int l = 0; l < NLEVELS; ++l) {
      float s  = (float)NL[l] * (1.0f / 512.0f);
      float ix = ((c.x * s + 1.0f) * 512.0f - 1.0f) * 0.5f;
      float iy = ((c.y * s + 1.0f) * 512.0f - 1.0f) * 0.5f;
      float x0f = floorf(ix), y0f = floorf(iy);
      float wx1 = ix - x0f, wy1 = iy - y0f;
      float wx0 = 1.0f - wx1, wy0 = 1.0f - wy1;
      int x0 = (int)x0f, y0 = (int)y0f;
      f32x2 a00, a01, a10, a11;
      if (l < KLDS) {
        // patch fully covers all taps for this level: no bounds checks
        const f32x2* p = (const f32x2*)((const char*)smem + LOFS[l]);
        int w   = PW[l];
        int idx = (y0 - P0[l]) * w + (x0 - P0[l]);
        a00 = p[idx];     a01 = p[idx + 1];
        a10 = p[idx + w]; a11 = p[idx + w + 1];
      } else if (l < NLEVELS - 1) {
        // interior: scale < 1 guarantees taps in [0,511]
        const f32x2* p = tbl + (size_t)l * PLANE + (y0 * 512 + x0);
        a00 = p[0];   a01 = p[1];
        a10 = p[512]; a11 = p[513];
      } else {
        // level 15: scale == 1.0, zeros padding at the border
        int x1 = x0 + 1, y1 = y0 + 1;
        wx0 *= (x0 >= 0)   ? 1.0f : 0.0f;
        wx1 *= (x1 <= 511) ? 1.0f : 0.0f;
        wy0 *= (y0 >= 0)   ? 1.0f : 0.0f;
        wy1 *= (y1 <= 511) ? 1.0f : 0.0f;
        int xc0 = min(max(x0, 0), 511), xc1 = min(x1, 511);
        int yc0 = min(max(y0, 0), 511), yc1 = min(y1, 511);
        const f32x2* p = tbl + (size_t)l * PLANE;
        a00 = p[yc0 * 512 + xc0]; a01 = p[yc0 * 512 + xc1];
        a10 = p[yc1 * 512 + xc0]; a11 = p[yc1 * 512 + xc1];
      }
      float w00 = wx0 * wy0, w01 = wx1 * wy0, w10 = wx0 * wy1, w11 = wx1 * wy1;
      acc0[l] = a00.x * w00 + a01.x * w01 + a10.x * w10 + a11.x * w11;
      acc1[l] = a00.y * w00 + a01.y * w01 + a10.y * w10 + a11.y * w11;
    }
    // out[b][f*16 + l]: 128B contiguous per point, streamed past L2
    f32x4* o = (f32x4*)(out + (size_t)b * 32);
    f32x4 v;
    v = (f32x4){acc0[0],  acc0[1],  acc0[2],  acc0[3]};  __builtin_nontemporal_store(v, o + 0);
    v = (f32x4){acc0[4],  acc0[5],  acc0[6],  acc0[7]};  __builtin_nontemporal_store(v, o + 1);
    v = (f32x4){acc0[8],  acc0[9],  acc0[10], acc0[11]}; __builtin_nontemporal_store(v, o + 2);
    v = (f32x4){acc0[12], acc0[13], acc0[14], acc0[15]}; __builtin_nontemporal_store(v, o + 3);
    v = (f32x4){acc1[0],  acc1[1],  acc1[2],  acc1[3]};  __builtin_nontemporal_store(v, o + 4);
    v = (f32x4){acc1[4],  acc1[5],  acc1[6],  acc1[7]};  __builtin_nontemporal_store(v, o + 5);
    v = (f32x4){acc1[8],  acc1[9],  acc1[10], acc1[11]}; __builtin_nontemporal_store(v, o + 6);
    v = (f32x4){acc1[12], acc1[13], acc1[14], acc1[15]}; __builtin_nontemporal_store(v, o + 7);
  }
}

// ---------------------------------------------------------------------------
// Fallback: no workspace -> sample directly from the original [l][f][y][x]
// ---------------------------------------------------------------------------
__global__ __launch_bounds__(256) void mrgrid_sample_direct(const f32x2* __restrict__ x,
                                                            const float* __restrict__ gt,
                                                            float* __restrict__ out,
                                                            int nPoints) {
  constexpr int NL[NLEVELS] = N_LIST;
  int stride = gridDim.x * blockDim.x;
  for (int b = blockIdx.x * blockDim.x + threadIdx.x; b < nPoints; b += stride) {
    f32x2 c = x[b];
    float acc0[NLEVELS], acc1[NLEVELS];
#pragma unroll
    for (int l = 0; l < NLEVELS; ++l) {
      float s  = (float)NL[l] * (1.0f / 512.0f);
      float ix = ((c.x * s + 1.0f) * 512.0f - 1.0f) * 0.5f;
      float iy = ((c.y * s + 1.0f) * 512.0f - 1.0f) * 0.5f;
      float x0f = floorf(ix), y0f = floorf(iy);
      float wx1 = ix - x0f, wy1 = iy - y0f;
      float wx0 = 1.0f - wx1, wy0 = 1.0f - wy1;
      int x0 = (int)x0f, y0 = (int)y0f;
      int x1 = x0 + 1, y1 = y0 + 1;
      wx0 *= (x0 >= 0)   ? 1.0f : 0.0f;
      wx1 *= (x1 <= 511) ? 1.0f : 0.0f;
      wy0 *= (y0 >= 0)   ? 1.0f : 0.0f;
      wy1 *= (y1 <= 511) ? 1.0f : 0.0f;
      int xc0 = min(max(x0, 0), 511), xc1 = min(x1, 511);
      int yc0 = min(max(y0, 0), 511), yc1 = min(y1, 511);
      const float* p0 = gt + (size_t)(2 * l) * PLANE;
      const float* p1 = p0 + PLANE;
      int i00 = yc0 * 512 + xc0, i01 = yc0 * 512 + xc1;
      int i10 = yc1 * 512 + xc0, i11 = yc1 * 512 + xc1;
      float w00 = wx0 * wy0, w01 = wx1 * wy0, w10 = wx0 * wy1, w11 = wx1 * wy1;
      acc0[l] = p0[i00] * w00 + p0[i01] * w01 + p0[i10] * w10 + p0[i11] * w11;
      acc1[l] = p1[i00] * w00 + p1[i01] * w01 + p1[i10] * w10 + p1[i11] * w11;
    }
    f32x4* o = (f32x4*)(out + (size_t)b * 32);
    f32x4 v;
    v = (f32x4){acc0[0],  acc0[1],  acc0[2],  acc0[3]};  __builtin_nontemporal_store(v, o + 0);
    v = (f32x4){acc0[4],  acc0[5],  acc0[6],  acc0[7]};  __builtin_nontemporal_store(v, o + 1);
    v = (f32x4){acc0[8],  acc0[9],  acc0[10], acc0[11]}; __builtin_nontemporal_store(v, o + 2);
    v = (f32x4){acc0[12], acc0[13], acc0[14], acc0[15]}; __builtin_nontemporal_store(v, o + 3);
    v = (f32x4){acc1[0],  acc1[1],  acc1[2],  acc1[3]};  __builtin_nontemporal_store(v, o + 4);
    v = (f32x4){acc1[4],  acc1[5],  acc1[6],  acc1[7]};  __builtin_nontemporal_store(v, o + 5);
    v = (f32x4){acc1[8],  acc1[9],  acc1[10], acc1[11]}; __builtin_nontemporal_store(v, o + 6);
    v = (f32x4){acc1[12], acc1[13], acc1[14], acc1[15]}; __builtin_nontemporal_store(v, o + 7);
  }
}

extern "C" void kernel_launch(void* const* d_in, const int* in_sizes, int n_in,
                              void* d_out, int out_size, void* d_ws, size_t ws_size,
                              hipStream_t stream) {
  const f32x2* x  = (const f32x2*)d_in[0];
  const float* gt = (const float*)d_in[1];
  float*       out = (float*)d_out;
  int nPoints = in_sizes[0] / 2;

  size_t tblBytes = (size_t)NLEVELS * PLANE * 2 * sizeof(float); // 32 MB
  if (ws_size >= tblBytes) {
    f32x2* tbl = (f32x2*)d_ws;
    int repackThreads = NLEVELS * PLANE;                 // 4,194,304
    mrgrid_repack<<<repackThreads / 256, 256, 0, stream>>>(gt, tbl);
    mrgrid_sample<<<1024, 256, SMEM_BYTES, stream>>>(x, tbl, out, nPoints);
  } else {
    mrgrid_sample_direct<<<2048, 256, 0, stream>>>(x, gt, out, nPoints);
  }
}